// ConvexContractionAttention_44220983279912
// MI455X (gfx1250) — compile-verified
//
#include <hip/hip_runtime.h>
#include <hip/hip_bf16.h>
#include <math.h>

#define D_CH   1024
#define C3     3072          // 3*D
#define NS     8192          // B*T = 4*2048
#define GAMMA_F   5.0f
#define EPS_NORM  1e-5f
#define EPS_W     1e-8f

#define STAT_CHUNKS 64       // x-stats: 64 chunks * 128 samples
#define MAIN_CHUNKS 256      // main pass: 256 chunks * 32 samples

typedef __attribute__((ext_vector_type(2))) float v2f;
typedef __attribute__((ext_vector_type(8))) float v8f;

// ---------------------------------------------------------------------------
// Kernel 1: per-channel sum & sum^2 of x over sample chunks, via WMMA f32
// 16x16x4 with an all-ones A matrix (D = 1^T * B + C accumulates column sums).
// One wave (32 lanes) per (16-channel tile, 128-sample chunk).
// ---------------------------------------------------------------------------
__global__ __launch_bounds__(32)
void k_stats_x(const float* __restrict__ x,
               float* __restrict__ psum, float* __restrict__ psq) {
    const int lane  = threadIdx.x;           // 0..31 (wave32)
    const int c0    = blockIdx.x * 16;       // channel tile base (0..3056)
    const int chunk = blockIdx.y;            // 0..STAT_CHUNKS-1
    const int ch    = c0 + (lane & 15);
    const int half  = lane >> 4;
    const int sbase = chunk * (NS / STAT_CHUNKS);   // 128 samples per chunk

    v2f ones = {1.0f, 1.0f};
    v8f csum = {};
    v8f csq  = {};

    // 32 quads of 4 samples; each lane supplies 2 distinct (sample,channel) elems.
    for (int q = 0; q < 32; ++q) {
        const int sb = sbase + q * 4;
        const float xa = x[(size_t)(sb + half)     * C3 + ch];
        const float xb = x[(size_t)(sb + 2 + half) * C3 + ch];
        v2f bsum = {xa, xb};
        v2f bsq  = {xa * xa, xb * xb};
        csum = __builtin_amdgcn_wmma_f32_16x16x4_f32(
            false, ones, false, bsum, (short)0, csum, false, false);
        csq  = __builtin_amdgcn_wmma_f32_16x16x4_f32(
            false, ones, false, bsq,  (short)0, csq,  false, false);
    }
    // Every row of D holds the column sums; lane L in [0,16): VGPR0 = (M=0, N=L).
    if (lane < 16) {
        psum[(size_t)chunk * C3 + c0 + lane] = csum[0];
        psq [(size_t)chunk * C3 + c0 + lane] = csq[0];
    }
}

// ---------------------------------------------------------------------------
// Kernel 2: deterministic serial reduction of chunk partials -> mean/var per
// channel (biased var, matching jnp.var default).
// ---------------------------------------------------------------------------
__global__ void k_reduce_x(const float* __restrict__ psum,
                           const float* __restrict__ psq,
                           float* __restrict__ meanx, float* __restrict__ varx) {
    const int c = blockIdx.x * blockDim.x + threadIdx.x;
    if (c >= C3) return;
    float s = 0.f, s2 = 0.f;
    for (int k = 0; k < STAT_CHUNKS; ++k) {
        s  += psum[(size_t)k * C3 + c];
        s2 += psq [(size_t)k * C3 + c];
    }
    const float m = s / (float)NS;
    meanx[c] = m;
    varx[c]  = s2 / (float)NS - m * m;
}

// ---------------------------------------------------------------------------
// Kernel 3: per-channel coefficient precompute. LAPACK-style Householder QR of
// the 3x3 'a', then A_j = g_j * c_j / sqrt(c_j^2*var + eps) with
// c_j = sum_i softplus(w_i) * Q[i][j].  Launched once per {q,k,v}.
// ---------------------------------------------------------------------------
__global__ void k_params(const float* __restrict__ w,   // (D,3)
                         const float* __restrict__ a,   // (D,3,3)
                         const float* __restrict__ g,   // (D,3)
                         const float* __restrict__ varx,// (D,) slice for this p
                         float* __restrict__ acoef) {   // (D,3) out
    const int d = blockIdx.x * blockDim.x + threadIdx.x;
    if (d >= D_CH) return;

    float A[3][3];
#pragma unroll
    for (int i = 0; i < 3; ++i)
#pragma unroll
        for (int j = 0; j < 3; ++j) A[i][j] = a[d * 9 + i * 3 + j];

    // Householder reflector 1 (LAPACK larfg sign convention).
    float tau0 = 0.f, v1 = 0.f, v2 = 0.f;
    {
        const float x0 = A[0][0], x1 = A[1][0], x2 = A[2][0];
        const float tail = x1 * x1 + x2 * x2;
        if (tail > 0.f) {
            const float nrm  = sqrtf(x0 * x0 + tail);
            const float beta = (x0 >= 0.f) ? -nrm : nrm;
            tau0 = (beta - x0) / beta;
            const float inv = 1.0f / (x0 - beta);
            v1 = x1 * inv; v2 = x2 * inv;
#pragma unroll
            for (int j = 0; j < 3; ++j) {
                float wv = A[0][j] + v1 * A[1][j] + v2 * A[2][j];
                wv *= tau0;
                A[0][j] -= wv; A[1][j] -= v1 * wv; A[2][j] -= v2 * wv;
            }
        }
    }
    // Householder reflector 2 on trailing 2x2.
    float tau1 = 0.f, u1 = 0.f;
    {
        const float y0 = A[1][1], y1 = A[2][1];
        if (y1 * y1 > 0.f) {
            const float nrm  = sqrtf(y0 * y0 + y1 * y1);
            const float beta = (y0 >= 0.f) ? -nrm : nrm;
            tau1 = (beta - y0) / beta;
            u1   = y1 / (y0 - beta);
        }
    }
    // Q = H1 * H2 * I   (3rd reflector is identity for a 3x3: tail length 0).
    float Q[3][3] = {{1.f,0.f,0.f},{0.f,1.f,0.f},{0.f,0.f,1.f}};
#pragma unroll
    for (int j = 0; j < 3; ++j) {      // apply H2 (rows 1,2)
        float wv = Q[1][j] + u1 * Q[2][j];
        wv *= tau1;
        Q[1][j] -= wv; Q[2][j] -= u1 * wv;
    }
#pragma unroll
    for (int j = 0; j < 3; ++j) {      // apply H1 (rows 0,1,2)
        float wv = Q[0][j] + v1 * Q[1][j] + v2 * Q[2][j];
        wv *= tau0;
        Q[0][j] -= wv; Q[1][j] -= v1 * wv; Q[2][j] -= v2 * wv;
    }

    const float var = varx[d];
#pragma unroll
    for (int j = 0; j < 3; ++j) {
        float c = 0.f;
#pragma unroll
        for (int i = 0; i < 3; ++i) {
            const float wi = w[d * 3 + i];
            const float sp = fmaxf(wi, 0.f) + log1pf(expf(-fabsf(wi))); // softplus
            c += sp * Q[i][j];
        }
        const float gj = g[d * 3 + j];
        acoef[d * 3 + j] = gj * c * rsqrtf(c * c * var + EPS_NORM);
    }
}

// ---------------------------------------------------------------------------
// Kernel 4: fused main pass. Each thread owns one channel d across a 32-sample
// chunk: computes Qp/Kp/Vp affinely, sigmoid-gated convex combination, writes
// staged output and accumulates per-d sum/sum^2 partials (deterministic).
// ---------------------------------------------------------------------------
__global__ __launch_bounds__(256)
void k_main(const float* __restrict__ x,
            const float* __restrict__ meanx,   // (3072,)
            const float* __restrict__ acoef,   // [3][D][3]
            const float* __restrict__ betaq,   // (D,3)
            const float* __restrict__ betak,
            const float* __restrict__ betav,
            float* __restrict__ outtmp,        // (NS, D)
            float* __restrict__ p2sum, float* __restrict__ p2sq) {
    const int d     = blockIdx.x * blockDim.x + threadIdx.x;   // 0..1023
    const int chunk = blockIdx.y;                              // 0..255
    const int s0    = chunk * (NS / MAIN_CHUNKS);              // 32 samples

    const float muq = meanx[d], muk = meanx[D_CH + d], muv = meanx[2 * D_CH + d];
    const float aq0 = acoef[d*3+0],             aq1 = acoef[d*3+1],             aq2 = acoef[d*3+2];
    const float ak0 = acoef[(D_CH+d)*3+0],      ak1 = acoef[(D_CH+d)*3+1],      ak2 = acoef[(D_CH+d)*3+2];
    const float av0 = acoef[(2*D_CH+d)*3+0],    av1 = acoef[(2*D_CH+d)*3+1],    av2 = acoef[(2*D_CH+d)*3+2];
    const float bq0 = betaq[d*3+0], bq1 = betaq[d*3+1], bq2 = betaq[d*3+2];
    const float bk0 = betak[d*3+0], bk1 = betak[d*3+1], bk2 = betak[d*3+2];
    const float bv0 = betav[d*3+0], bv1 = betav[d*3+1], bv2 = betav[d*3+2];

    float acc = 0.f, acc2 = 0.f;
    for (int s = s0; s < s0 + NS / MAIN_CHUNKS; ++s) {
        const size_t base = (size_t)s * C3;
        __builtin_prefetch(x + base + C3 + d, 0, 0);   // global_prefetch_b8
        const float xq = x[base + d]            - muq;
        const float xk = x[base + D_CH + d]     - muk;
        const float xv = x[base + 2 * D_CH + d] - muv;

        const float q0 = fmaf(aq0, xq, bq0), q1 = fmaf(aq1, xq, bq1), q2 = fmaf(aq2, xq, bq2);
        const float k0 = fmaf(ak0, xk, bk0), k1 = fmaf(ak1, xk, bk1), k2 = fmaf(ak2, xk, bk2);
        const float w0 = fmaf(av0, xv, bv0), w1 = fmaf(av1, xv, bv1), w2 = fmaf(av2, xv, bv2);

        const float s0g = 1.0f / (1.0f + expf(-GAMMA_F * q0 * k0));
        const float s1g = 1.0f / (1.0f + expf(-GAMMA_F * q1 * k1));
        const float s2g = 1.0f / (1.0f + expf(-GAMMA_F * q2 * k2));
        const float den = s0g + s1g + s2g + EPS_W;
        const float o   = (s0g * w0 + s1g * w1 + s2g * w2) / den;

        outtmp[(size_t)s * D_CH + d] = o;
        acc += o; acc2 += o * o;
    }
    p2sum[(size_t)chunk * D_CH + d] = acc;
    p2sq [(size_t)chunk * D_CH + d] = acc2;
}

// ---------------------------------------------------------------------------
// Kernel 5: reduce out-partials -> per-d mean/var.
// ---------------------------------------------------------------------------
__global__ void k_reduce_o(const float* __restrict__ p2sum,
                           const float* __restrict__ p2sq,
                           float* __restrict__ meano, float* __restrict__ varo) {
    const int d = blockIdx.x * blockDim.x + threadIdx.x;
    if (d >= D_CH) return;
    float s = 0.f, s2 = 0.f;
    for (int k = 0; k < MAIN_CHUNKS; ++k) {
        s  += p2sum[(size_t)k * D_CH + d];
        s2 += p2sq [(size_t)k * D_CH + d];
    }
    const float m = s / (float)NS;
    meano[d] = m;
    varo[d]  = s2 / (float)NS - m * m;
}

// ---------------------------------------------------------------------------
// Kernel 6: final batch affine norm.
// ---------------------------------------------------------------------------
__global__ __launch_bounds__(256)
void k_final(const float* __restrict__ outtmp,
             const float* __restrict__ meano, const float* __restrict__ varo,
             const float* __restrict__ g_out, const float* __restrict__ b_out,
             float* __restrict__ out) {
    const size_t i = (size_t)blockIdx.x * blockDim.x + threadIdx.x;
    if (i >= (size_t)NS * D_CH) return;
    const int d = (int)(i & (D_CH - 1));
    const float inv = rsqrtf(varo[d] + EPS_NORM);
    out[i] = (outtmp[i] - meano[d]) * inv * g_out[d] + b_out[d];
}

// ---------------------------------------------------------------------------
extern "C" void kernel_launch(void* const* d_in, const int* in_sizes, int n_in,
                              void* d_out, int out_size, void* d_ws, size_t ws_size,
                              hipStream_t stream) {
    const float* x     = (const float*)d_in[0];
    const float* wq    = (const float*)d_in[1];
    const float* aq    = (const float*)d_in[3];
    const float* gq    = (const float*)d_in[4];
    const float* betaq = (const float*)d_in[5];
    const float* wk    = (const float*)d_in[6];
    const float* ak    = (const float*)d_in[8];
    const float* gk    = (const float*)d_in[9];
    const float* betak = (const float*)d_in[10];
    const float* wv    = (const float*)d_in[11];
    const float* av    = (const float*)d_in[13];
    const float* gv    = (const float*)d_in[14];
    const float* betav = (const float*)d_in[15];
    const float* g_out = (const float*)d_in[16];
    const float* b_out = (const float*)d_in[17];

    float* W = (float*)d_ws;
    float* OUTTMP = W;                                   // 8,388,608
    float* P1SUM  = OUTTMP + (size_t)NS * D_CH;          //   196,608
    float* P1SQ   = P1SUM  + (size_t)STAT_CHUNKS * C3;   //   196,608
    float* MEANX  = P1SQ   + (size_t)STAT_CHUNKS * C3;   //     3,072
    float* VARX   = MEANX  + C3;                         //     3,072
    float* ACOEF  = VARX   + C3;                         //     9,216
    float* P2SUM  = ACOEF  + 3 * D_CH * 3;               //   262,144
    float* P2SQ   = P2SUM  + (size_t)MAIN_CHUNKS * D_CH; //   262,144
    float* MEANO  = P2SQ   + (size_t)MAIN_CHUNKS * D_CH; //     1,024
    float* VARO   = MEANO  + D_CH;                       //     1,024

    // 1) x statistics via WMMA column-sum reduction.
    k_stats_x<<<dim3(C3 / 16, STAT_CHUNKS), 32, 0, stream>>>(x, P1SUM, P1SQ);
    k_reduce_x<<<(C3 + 255) / 256, 256, 0, stream>>>(P1SUM, P1SQ, MEANX, VARX);

    // 2) per-channel affine coefficients (QR + softplus folded into A).
    k_params<<<(D_CH + 255) / 256, 256, 0, stream>>>(wq, aq, gq, VARX,            ACOEF);
    k_params<<<(D_CH + 255) / 256, 256, 0, stream>>>(wk, ak, gk, VARX + D_CH,     ACOEF + D_CH * 3);
    k_params<<<(D_CH + 255) / 256, 256, 0, stream>>>(wv, av, gv, VARX + 2 * D_CH, ACOEF + 2 * D_CH * 3);

    // 3) fused main pass + out-stat partials.
    k_main<<<dim3(D_CH / 256, MAIN_CHUNKS), 256, 0, stream>>>(
        x, MEANX, ACOEF, betaq, betak, betav, OUTTMP, P2SUM, P2SQ);
    k_reduce_o<<<(D_CH + 255) / 256, 256, 0, stream>>>(P2SUM, P2SQ, MEANO, VARO);

    // 4) final batch affine norm.
    k_final<<<(unsigned)(((size_t)NS * D_CH) / 256), 256, 0, stream>>>(
        OUTTMP, MEANO, VARO, g_out, b_out, (float*)d_out);
}